// BallConvolution_66116726554757
// MI455X (gfx1250) — compile-verified
//
#include <hip/hip_runtime.h>

typedef float v2f __attribute__((ext_vector_type(2)));
typedef float v8f __attribute__((ext_vector_type(8)));

#define N_NODES  100000
#define N_ETYPES 3
#define N_EDGES  1600000
#define D_IN     128
#define D_HID    256
#define D_OUT    32

// ---------------------------------------------------------------- fill
__global__ void fill_zero_kernel(float* __restrict__ p, int n) {
    int i = blockIdx.x * blockDim.x + threadIdx.x;
    if (i < n) p[i] = 0.0f;
}

// ---------------------------------------------------------------- degrees
// dst: [N_ETYPES][N_EDGES], deg: [N_ETYPES][N_NODES] (float counts)
__global__ void degree_kernel(const int* __restrict__ dst, float* __restrict__ deg) {
    int e = blockIdx.y;
    int i = blockIdx.x * blockDim.x + threadIdx.x;
    if (i < N_EDGES) {
        int d = dst[(size_t)e * N_EDGES + i];
        atomicAdd(&deg[(size_t)e * N_NODES + d], 1.0f);
    }
}

// ---------------------------------------------------------------- gather + scatter-add
// 256 threads = 8 edges/block; 32 lanes/edge, float4 per lane (128 feats).
__global__ void scatter_kernel(const float* __restrict__ h,
                               const int* __restrict__ src,
                               const int* __restrict__ dst,
                               float* __restrict__ agg) {
    int edge = blockIdx.x * 8 + (threadIdx.x >> 5);
    int c    = (threadIdx.x & 31) * 4;
    if (edge >= N_EDGES) return;
    int s = src[edge];
    int d = dst[edge];
    const float4 v = *(const float4*)(h + (size_t)s * D_IN + c);
    float* o = agg + (size_t)d * D_IN + c;
    atomicAdd(o + 0, v.x);
    atomicAdd(o + 1, v.y);
    atomicAdd(o + 2, v.z);
    atomicAdd(o + 3, v.w);
}

// ---------------------------------------------------------------- fp32 WMMA GEMM
// One wave32 per 16x16 output tile of C[M x N] = rowscale(A[M x K]) @ W[K x N] + bias.
// Optional per-row scale from degree (1/max(deg,1)), optional ReLU, optional
// accumulate (C += outScale*val) — used for the 1/3 etype-mean fusion.
__global__ void wmma_gemm_f32_kernel(const float* __restrict__ A,
                                     const float* __restrict__ W,
                                     const float* __restrict__ bias,
                                     float* __restrict__ C,
                                     const float* __restrict__ deg,  // [M] or null
                                     int M, int N, int K,
                                     float outScale, int accumulate, int doRelu) {
    const int lane = threadIdx.x;        // 0..31, full wave (EXEC all ones)
    const int half = lane >> 4;          // 0 or 1
    const int l15  = lane & 15;
    const int n0   = blockIdx.x * 16;
    const int m0   = blockIdx.y * 16;
    const int rowA = m0 + l15;           // A-fragment row for this lane
    const int colB = n0 + l15;           // B/C-fragment column for this lane

    float rs = 1.0f;
    if (deg) rs = 1.0f / fmaxf(deg[rowA], 1.0f);

    v8f acc = {};
    const float* arow = A + (size_t)rowA * K;
    for (int k = 0; k < K; k += 4) {
        const int ka = k + half * 2;     // lanes 16-31 carry K+2, K+3
        v2f a, b;
        a.x = arow[ka]     * rs;
        a.y = arow[ka + 1] * rs;
        b.x = W[(size_t)ka       * N + colB];
        b.y = W[(size_t)(ka + 1) * N + colB];
        acc = __builtin_amdgcn_wmma_f32_16x16x4_f32(
            /*neg_a=*/false, a, /*neg_b=*/false, b,
            /*c_mod=*/(short)0, acc, /*reuse_a=*/false, /*reuse_b=*/false);
    }

    const float bv = bias[colB];
#pragma unroll
    for (int r = 0; r < 8; ++r) {
        const int row = m0 + r + half * 8;     // C layout: VGPR r -> M = r (+8 for hi half)
        float v = acc[r] + bv;
        if (doRelu) v = fmaxf(v, 0.0f);
        v *= outScale;
        const size_t idx = (size_t)row * N + colB;
        if (accumulate) C[idx] += v;           // wave owns tile: race-free RMW
        else            C[idx] = v;
    }
}

// ---------------------------------------------------------------- driver
extern "C" void kernel_launch(void* const* d_in, const int* in_sizes, int n_in,
                              void* d_out, int out_size, void* d_ws, size_t ws_size,
                              hipStream_t stream) {
    const float* blocks = (const float*)d_in[0];
    const int*   esrc   = (const int*)  d_in[1];
    const int*   edst   = (const int*)  d_in[2];
    const float* convW  = (const float*)d_in[3];
    const float* convB  = (const float*)d_in[4];
    const float* W1     = (const float*)d_in[5];
    const float* b1     = (const float*)d_in[6];
    const float* W2     = (const float*)d_in[7];
    const float* b2     = (const float*)d_in[8];
    float*       out    = (float*)d_out;

    char* ws = (char*)d_ws;
    const size_t BUF = (size_t)N_NODES * D_IN * sizeof(float);   // 51.2 MB
    float* buf0 = (float*)(ws);                // layer-0 output / later H1 (spans buf0+agg)
    float* agg  = (float*)(ws + BUF);          // per-etype aggregation scratch
    float* buf2 = (float*)(ws + 2 * BUF);      // layer-1 output
    float* deg  = (float*)(ws + 3 * BUF);      // [N_ETYPES][N_NODES]
    float* H1   = buf0;                        // 100k x 256 reuses buf0+agg after convs

    // in-degrees per etype (dst is the same for both layers)
    fill_zero_kernel<<<(N_ETYPES * N_NODES + 255) / 256, 256, 0, stream>>>(deg, N_ETYPES * N_NODES);
    {
        dim3 g((N_EDGES + 255) / 256, N_ETYPES);
        degree_kernel<<<g, 256, 0, stream>>>(edst, deg);
    }

    const float* hin = blocks;
    float* hout = buf0;
    for (int l = 0; l < 2; ++l) {
        for (int e = 0; e < N_ETYPES; ++e) {
            fill_zero_kernel<<<(N_NODES * D_IN + 255) / 256, 256, 0, stream>>>(agg, N_NODES * D_IN);
            scatter_kernel<<<(N_EDGES + 7) / 8, 256, 0, stream>>>(
                hin, esrc + (size_t)e * N_EDGES, edst + (size_t)e * N_EDGES, agg);
            dim3 g(D_IN / 16, N_NODES / 16);
            wmma_gemm_f32_kernel<<<g, 32, 0, stream>>>(
                agg,
                convW + ((size_t)l * N_ETYPES + e) * D_IN * D_IN,
                convB + ((size_t)l * N_ETYPES + e) * D_IN,
                hout,
                deg + (size_t)e * N_NODES,
                N_NODES, D_IN, D_IN,
                1.0f / 3.0f, /*accumulate=*/(e > 0) ? 1 : 0, /*relu=*/1);
        }
        hin  = hout;      // after l=0: buf0; after l=1: buf2
        hout = buf2;
    }

    // classifier head (no ReLU between the two linears, per reference)
    {
        dim3 g(D_HID / 16, N_NODES / 16);
        wmma_gemm_f32_kernel<<<g, 32, 0, stream>>>(
            hin, W1, b1, H1, nullptr, N_NODES, D_HID, D_IN, 1.0f, 0, 0);
    }
    {
        dim3 g(D_OUT / 16, N_NODES / 16);
        wmma_gemm_f32_kernel<<<g, 32, 0, stream>>>(
            H1, W2, b2, out, nullptr, N_NODES, D_OUT, D_HID, 1.0f, 0, 0);
    }
}